// SpanClassificationMarginLoss_17222818857327
// MI455X (gfx1250) — compile-verified
//
#include <hip/hip_runtime.h>
#include <cstdint>

#define BATCH 64
#define SEQ   128
#define LBL   32
#define NEGINF (-1000000000.0f)

// ---------------------------------------------------------------------------
// Kernel A: one block per (b, i) row. Async-stage 16KB of logits into LDS,
// compute span potentials pot[b,i,j], sequence lengths (i==0 rows), and the
// per-row gold-score partial sum (deterministic tree reduction).
// ---------------------------------------------------------------------------
__global__ __launch_bounds__(128) void pot_gold_kernel(
    const float* __restrict__ logits, const int* __restrict__ labels,
    float* __restrict__ pot, float* __restrict__ goldPart, int* __restrict__ lenB)
{
    __shared__ float xbuf[SEQ * LBL];   // 16 KB: logits[b][i][0:128][0:32]
    __shared__ float fred[128];
    __shared__ int   ired[128];

    const int bx  = blockIdx.x;         // = b*SEQ + i
    const int i   = bx & (SEQ - 1);
    const int tid = threadIdx.x;        // = j

    // ---- CDNA5 async global->LDS staging (ASYNCcnt-tracked) ---------------
    const float* rowp = logits + (size_t)bx * (SEQ * LBL);
    const unsigned ldsbase = (unsigned)(uintptr_t)(&xbuf[0]);
    const unsigned long long gbase = (unsigned long long)(uintptr_t)rowp;
    #pragma unroll
    for (int r = 0; r < 8; ++r) {
        unsigned off = (unsigned)((r * 128 + tid) * 16);
        unsigned lds = ldsbase + off;
        asm volatile("global_load_async_to_lds_b128 %0, %1, %2"
                     :: "v"(lds), "v"(off), "s"(gbase) : "memory");
    }

    // Label for this (b,i,j); gold uses max(label,0)
    const int labRaw = labels[(size_t)bx * SEQ + tid];
    int labc = labRaw < 0 ? 0 : labRaw;
    labc = labc > (LBL - 1) ? (LBL - 1) : labc;

    // Sequence length: only i==0 rows compute count(labels[b,0,:] != -100)
    int slen = SEQ;
    if (i == 0) {
        ired[tid] = (labRaw != -100) ? 1 : 0;
        __syncthreads();
        for (int s = 64; s > 0; s >>= 1) {
            if (tid < s) ired[tid] += ired[tid + s];
            __syncthreads();
        }
        slen = ired[0];
        if (tid == 0) lenB[bx >> 7] = slen;
    }

    // Wait for the async copies from THIS wave, then make LDS block-visible
    asm volatile("s_wait_asynccnt 0" ::: "memory");
    __syncthreads();

    // pot[b,i,j] = max_l (x[l] + aug[l]);  gold partial = x[label]
    float xs[LBL];
    #pragma unroll
    for (int c = 0; c < 8; ++c) {
        float4 q = *reinterpret_cast<const float4*>(&xbuf[tid * LBL + c * 4]);
        xs[c * 4 + 0] = q.x; xs[c * 4 + 1] = q.y;
        xs[c * 4 + 2] = q.z; xs[c * 4 + 3] = q.w;
    }
    const float x0 = xs[0];
    const float extra0 = (i == 0 && tid == slen - 1) ? NEGINF : 0.0f;
    float m = NEGINF, gold = 0.0f;
    #pragma unroll
    for (int l = 0; l < LBL; ++l) {
        float d = xs[l] - x0;
        gold = (l == labc) ? d : gold;
        float a = (l == labc) ? 0.0f : 1.0f;
        if (l == 0) a += extra0;            // augment[b,0,len-1,0] += -1e9
        m = fmaxf(m, d + a);
    }
    pot[(size_t)bx * SEQ + tid] = m;

    fred[tid] = gold;
    __syncthreads();
    for (int s = 64; s > 0; s >>= 1) {
        if (tid < s) fred[tid] += fred[tid + s];
        __syncthreads();
    }
    if (tid == 0) goldPart[bx] = fred[0];
}

// ---------------------------------------------------------------------------
// Kernel B: one block per batch element. Max-plus CKY inside pass entirely in
// LDS. bt[w][i] = best score of span [i, i+w] (transposed chart, padded).
// Adaptive threads-per-span with intra-wave shfl_xor max-reduction.
// ---------------------------------------------------------------------------
__global__ __launch_bounds__(128) void cky_kernel(
    const float* __restrict__ pot, const float* __restrict__ goldPart,
    const int* __restrict__ lenB, float* __restrict__ margins)
{
    __shared__ float bt[SEQ][SEQ + 2];  // 66,560 B, stride 130 (bank-friendly)
    __shared__ float ps[SEQ][SEQ];      // 65,536 B staged pot chart
    __shared__ float fred[128];

    const int b = blockIdx.x, tid = threadIdx.x;

    // Async-stage pot[b] (64 KB) into LDS
    const float* pb = pot + (size_t)b * SEQ * SEQ;
    const unsigned ldsbase = (unsigned)(uintptr_t)(&ps[0][0]);
    const unsigned long long gbase = (unsigned long long)(uintptr_t)pb;
    for (int r = 0; r < 32; ++r) {
        unsigned off = (unsigned)((r * 128 + tid) * 16);
        unsigned lds = ldsbase + off;
        asm volatile("global_load_async_to_lds_b128 %0, %1, %2"
                     :: "v"(lds), "v"(off), "s"(gbase) : "memory");
    }

    // Overlap: reduce the 128 gold partials for this batch (deterministic)
    fred[tid] = goldPart[b * SEQ + tid];
    __syncthreads();
    for (int s = 64; s > 0; s >>= 1) {
        if (tid < s) fred[tid] += fred[tid + s];
        __syncthreads();
    }

    asm volatile("s_wait_asynccnt 0" ::: "memory");
    __syncthreads();

    // Width-0 spans (terminals): bt[0][i] = pot[i][i]
    bt[0][tid] = ps[tid][tid];
    __syncthreads();

    for (int w = 1; w < SEQ; ++w) {
        const int spans = SEQ - w;
        int t = 1;                                   // threads per span
        while (t < 32 && (t << 1) * spans <= 128) t <<= 1;
        const int span = tid / t;
        const int sub  = tid % t;                    // contiguous -> same wave
        float m = NEGINF;
        if (span < spans) {
            for (int k = sub; k < w; k += t)
                m = fmaxf(m, bt[k][span] + bt[w - 1 - k][span + k + 1]);
        }
        for (int d = t >> 1; d > 0; d >>= 1)         // intra-wave max-reduce
            m = fmaxf(m, __shfl_xor(m, d, 32));
        if (span < spans && sub == 0)
            bt[w][span] = m + ps[span][span + w];    // row w unread this iter
        __syncthreads();
    }

    if (tid == 0) {
        const int len = lenB[b];
        const float pred = bt[len - 1][0];
        margins[b] = fmaxf(pred - fred[0], 0.0f);    // relu(pred - gold)
    }
}

// ---------------------------------------------------------------------------
// Kernel C: mean of 64 margins -> scalar loss
// ---------------------------------------------------------------------------
__global__ __launch_bounds__(64) void finalize_kernel(
    const float* __restrict__ margins, float* __restrict__ out)
{
    __shared__ float r[64];
    const int tid = threadIdx.x;
    r[tid] = margins[tid];
    __syncthreads();
    for (int s = 32; s > 0; s >>= 1) {
        if (tid < s) r[tid] += r[tid + s];
        __syncthreads();
    }
    if (tid == 0) out[0] = r[0] * (1.0f / BATCH);
}

// ---------------------------------------------------------------------------
extern "C" void kernel_launch(void* const* d_in, const int* in_sizes, int n_in,
                              void* d_out, int out_size, void* d_ws, size_t ws_size,
                              hipStream_t stream) {
    (void)in_sizes; (void)n_in; (void)out_size; (void)ws_size;
    const float* logits = (const float*)d_in[0];
    const int*   labels = (const int*)d_in[1];

    float* pot      = (float*)d_ws;                       // B*N*N  (4 MB)
    float* goldPart = pot + (size_t)BATCH * SEQ * SEQ;    // B*N
    float* margins  = goldPart + BATCH * SEQ;             // B
    int*   lenB     = (int*)(margins + BATCH);            // B

    pot_gold_kernel<<<dim3(BATCH * SEQ), dim3(128), 0, stream>>>(
        logits, labels, pot, goldPart, lenB);
    cky_kernel<<<dim3(BATCH), dim3(128), 0, stream>>>(
        pot, goldPart, lenB, margins);
    finalize_kernel<<<dim3(1), dim3(64), 0, stream>>>(
        margins, (float*)d_out);
}